// ICP_64372969832536
// MI455X (gfx1250) — compile-verified
//
#include <hip/hip_runtime.h>
#include <hip/hip_bf16.h>

typedef float f32x2 __attribute__((ext_vector_type(2)));
typedef float f32x8 __attribute__((ext_vector_type(8)));

#define TOL 1e-4f
#define STEPS 11   // STEPLIM + 1 scan steps

// ---------------------------------------------------------------------------
// prep: pack p1 -> T (temppc, float4 xyz0), P1 (copy of p1), p2 -> P with
// w = -0.5*|p|^2 (the fused-bias trick for the WMMA K=4 slot). init state.
// ---------------------------------------------------------------------------
__global__ void icp_prep(const float* __restrict__ p1, const float* __restrict__ p2,
                         float4* __restrict__ T, float4* __restrict__ P1,
                         float4* __restrict__ P, float* __restrict__ state, int N) {
    int i = blockIdx.x * blockDim.x + threadIdx.x;
    if (i < N) {
        float x = p1[3*i], y = p1[3*i+1], z = p1[3*i+2];
        T[i]  = make_float4(x, y, z, 0.f);
        P1[i] = make_float4(x, y, z, 0.f);
        float a = p2[3*i], b = p2[3*i+1], c = p2[3*i+2];
        P[i] = make_float4(a, b, c, -0.5f * (a*a + b*b + c*c));
    }
    if (i == 0) { state[0] = __builtin_inff(); state[1] = 0.f; }
}

// ---------------------------------------------------------------------------
// 1-NN via V_WMMA_F32_16X16X4_F32.
// A (16x4): rows = 16 temppc points, cols = (x, y, z, 1)
// B (4x16): cols = 16 p2 points,   rows = (x, y, z, -0.5*|p|^2)
// D[m][n] = dot(t_m, p_n) - 0.5*|p_n|^2  ->  argmin d2 == argmax D.
// A layout (32-bit 16x4): V0: lanes0-15 K=0, lanes16-31 K=2; V1: K=1 / K=3.
// B layout (32-bit 4x16): V0: lanes0-15 K=0, lanes16-31 K=2; V1: K=1 / K=3.
// C layout: VGPR v holds row M=v (lanes 0-15) / M=v+8 (lanes 16-31), N=lane%16.
// One wave handles a 16-row tile over all N columns.
// ---------------------------------------------------------------------------
__global__ void icp_knn(const float4* __restrict__ T, const float4* __restrict__ P,
                        int* __restrict__ outIdx, float* __restrict__ outD, int N) {
    int wave = (blockIdx.x * blockDim.x + threadIdx.x) >> 5;
    int lane = threadIdx.x & 31;
    int rowBase = wave * 16;
    if (rowBase >= N) return;

    bool hi = lane >= 16;
    int sub = lane & 15;

    float4 tp = T[rowBase + sub];
    f32x2 a;
    a.x = hi ? tp.z : tp.x;   // K=2 : K=0
    a.y = hi ? 1.0f : tp.y;   // K=3 : K=1

    f32x8 bestv;
    int   besti[8];
#pragma unroll
    for (int v = 0; v < 8; ++v) { bestv[v] = -3.0e38f; besti[v] = 0; }

#pragma unroll 4
    for (int j = 0; j < N; j += 16) {
        float4 pp = P[j + sub];
        f32x2 b;
        b.x = hi ? pp.z : pp.x;   // K=2 : K=0
        b.y = hi ? pp.w : pp.y;   // K=3 (=-0.5|p|^2) : K=1
        f32x8 c = {};
        c = __builtin_amdgcn_wmma_f32_16x16x4_f32(
                false, a, false, b, (short)0, c, false, false);
#pragma unroll
        for (int v = 0; v < 8; ++v) {
            float s = c[v];
            if (s > bestv[v]) { bestv[v] = s; besti[v] = j + sub; }
        }
    }

    // butterfly argmax across the 16 lanes holding each row (tie -> lowest idx)
#pragma unroll
    for (int m = 8; m >= 1; m >>= 1) {
#pragma unroll
        for (int v = 0; v < 8; ++v) {
            float ov = __shfl_xor(bestv[v], m, 32);
            int   oi = __shfl_xor(besti[v], m, 32);
            if (ov > bestv[v] || (ov == bestv[v] && oi < besti[v])) {
                bestv[v] = ov; besti[v] = oi;
            }
        }
    }

    if (sub == 0) {
#pragma unroll
        for (int v = 0; v < 8; ++v) {
            int row = rowBase + v + (hi ? 8 : 0);
            float4 q = T[row];
            float sq1 = q.x*q.x + q.y*q.y + q.z*q.z;
            float d2  = sq1 - 2.0f * bestv[v];
            outIdx[row] = besti[v];
            outD[row]   = sqrtf(fmaxf(d2, 0.0f));
        }
    }
}

// ---------------------------------------------------------------------------
// single-block 16-way reduction:
// sums[0..2]=sum T, [3..5]=sum matched, [6..14]=sum T_i (outer) M_i, [15]=sum dmin
// idx == nullptr -> matched = M[i] (identity pairing, used for the final fit)
// ---------------------------------------------------------------------------
__global__ void icp_reduce(const float4* __restrict__ Tm, const float4* __restrict__ Mm,
                           const int* __restrict__ idx, const float* __restrict__ dv,
                           float* __restrict__ sums, int N) {
    __shared__ float sm[256];
    float acc[16];
#pragma unroll
    for (int q = 0; q < 16; ++q) acc[q] = 0.f;

    for (int i = threadIdx.x; i < N; i += 256) {
        float4 t = Tm[i];
        float4 m = Mm[idx ? idx[i] : i];
        acc[0] += t.x; acc[1] += t.y; acc[2] += t.z;
        acc[3] += m.x; acc[4] += m.y; acc[5] += m.z;
        acc[6]  += t.x*m.x; acc[7]  += t.x*m.y; acc[8]  += t.x*m.z;
        acc[9]  += t.y*m.x; acc[10] += t.y*m.y; acc[11] += t.y*m.z;
        acc[12] += t.z*m.x; acc[13] += t.z*m.y; acc[14] += t.z*m.z;
        if (dv) acc[15] += dv[i];
    }
    for (int q = 0; q < 16; ++q) {
        sm[threadIdx.x] = acc[q];
        __syncthreads();
        for (int off = 128; off > 0; off >>= 1) {
            if (threadIdx.x < off) sm[threadIdx.x] += sm[threadIdx.x + off];
            __syncthreads();
        }
        if (threadIdx.x == 0) sums[q] = sm[0];
        __syncthreads();
    }
}

// ---------------------------------------------------------------------------
// 3x3 Kabsch via Jacobi eigendecomposition of H^T H (single thread)
// ---------------------------------------------------------------------------
__device__ float icp_det3(const float M[3][3]) {
    return M[0][0]*(M[1][1]*M[2][2]-M[1][2]*M[2][1])
         - M[0][1]*(M[1][0]*M[2][2]-M[1][2]*M[2][0])
         + M[0][2]*(M[1][0]*M[2][1]-M[1][1]*M[2][0]);
}

__device__ void icp_kabsch(const float* s, float Nf, float R[3][3], float t[3]) {
    float c1[3] = { s[0]/Nf, s[1]/Nf, s[2]/Nf };
    float c2[3] = { s[3]/Nf, s[4]/Nf, s[5]/Nf };
    float H[3][3];
    for (int i = 0; i < 3; ++i)
        for (int j = 0; j < 3; ++j)
            H[i][j] = s[6 + 3*i + j] - Nf * c1[i] * c2[j];

    // A = H^T H (symmetric PSD)
    float A[3][3];
    for (int i = 0; i < 3; ++i)
        for (int j = 0; j < 3; ++j) {
            float v = 0.f;
            for (int k = 0; k < 3; ++k) v += H[k][i] * H[k][j];
            A[i][j] = v;
        }
    float V[3][3] = {{1,0,0},{0,1,0},{0,0,1}};

    const int PQ[3][2] = {{0,1},{0,2},{1,2}};
    for (int sweep = 0; sweep < 25; ++sweep) {
        for (int r = 0; r < 3; ++r) {
            int p = PQ[r][0], q = PQ[r][1];
            float apq = A[p][q];
            if (fabsf(apq) < 1e-22f) continue;
            float theta = (A[q][q] - A[p][p]) / (2.0f * apq);
            float tt = copysignf(1.0f, theta) / (fabsf(theta) + sqrtf(theta*theta + 1.0f));
            float cc = 1.0f / sqrtf(tt*tt + 1.0f);
            float ss = tt * cc;
            for (int k = 0; k < 3; ++k) {
                float akp = A[k][p], akq = A[k][q];
                A[k][p] = cc*akp - ss*akq;
                A[k][q] = ss*akp + cc*akq;
            }
            for (int k = 0; k < 3; ++k) {
                float apk = A[p][k], aqk = A[q][k];
                A[p][k] = cc*apk - ss*aqk;
                A[q][k] = ss*apk + cc*aqk;
            }
            for (int k = 0; k < 3; ++k) {
                float vkp = V[k][p], vkq = V[k][q];
                V[k][p] = cc*vkp - ss*vkq;
                V[k][q] = ss*vkp + cc*vkq;
            }
        }
    }
    // sort eigenvalues descending (column swaps)
    float lam[3] = { A[0][0], A[1][1], A[2][2] };
    for (int i = 0; i < 2; ++i)
        for (int j = i + 1; j < 3; ++j)
            if (lam[j] > lam[i]) {
                float tl = lam[i]; lam[i] = lam[j]; lam[j] = tl;
                for (int k = 0; k < 3; ++k) {
                    float tv = V[k][i]; V[k][i] = V[k][j]; V[k][j] = tv;
                }
            }
    // U columns: u_k = H v_k / s_k
    float U[3][3];
    for (int k = 0; k < 3; ++k) {
        float u[3];
        for (int i = 0; i < 3; ++i)
            u[i] = H[i][0]*V[0][k] + H[i][1]*V[1][k] + H[i][2]*V[2][k];
        float n2 = u[0]*u[0] + u[1]*u[1] + u[2]*u[2];
        if (n2 > 1e-24f) {
            float inv = 1.0f / sqrtf(n2);
            for (int i = 0; i < 3; ++i) U[i][k] = u[i] * inv;
        } else {
            // rank-deficient: third direction = cross of first two
            U[0][k] = U[1][(k+1)%3]*U[2][(k+2)%3] - U[2][(k+1)%3]*U[1][(k+2)%3];
            U[1][k] = U[2][(k+1)%3]*U[0][(k+2)%3] - U[0][(k+1)%3]*U[2][(k+2)%3];
            U[2][k] = U[0][(k+1)%3]*U[1][(k+2)%3] - U[1][(k+1)%3]*U[0][(k+2)%3];
        }
    }
    float d = copysignf(1.0f, icp_det3(V) * icp_det3(U));  // sign(det(V U^T))
    for (int i = 0; i < 3; ++i)
        for (int j = 0; j < 3; ++j)
            R[i][j] = V[i][0]*U[j][0] + V[i][1]*U[j][1] + d * V[i][2]*U[j][2];
    for (int i = 0; i < 3; ++i)
        t[i] = c2[i] - (R[i][0]*c1[0] + R[i][1]*c1[1] + R[i][2]*c1[2]);
}

// state: [0]=err, [1]=done, [2..10]=R, [11..13]=t
__global__ void icp_solve_step(const float* __restrict__ sums, float* __restrict__ state, int N) {
    if (threadIdx.x != 0 || blockIdx.x != 0) return;
    float Nf = (float)N;
    float errnew = sums[15] / Nf;
    float err = state[0], done = state[1];
    float rel = fabsf((errnew - err) / err);          // inf->NaN on first iter: NaN < TOL == false
    int new_done = (done != 0.f) || (rel < TOL);
    float R[3][3], t[3];
    icp_kabsch(sums, Nf, R, t);
    state[1] = new_done ? 1.f : 0.f;
    if (!new_done) state[0] = errnew;
    for (int i = 0; i < 3; ++i)
        for (int j = 0; j < 3; ++j)
            state[2 + 3*i + j] = R[i][j];
    state[11] = t[0]; state[12] = t[1]; state[13] = t[2];
}

__global__ void icp_apply(float4* __restrict__ T, const float* __restrict__ state, int N) {
    int i = blockIdx.x * blockDim.x + threadIdx.x;
    if (i >= N) return;
    if (state[1] != 0.f) return;                      // done -> freeze temppc
    const float* R = state + 2;
    const float* t = state + 11;
    float4 p = T[i];
    float x = R[0]*p.x + R[1]*p.y + R[2]*p.z + t[0];
    float y = R[3]*p.x + R[4]*p.y + R[5]*p.z + t[1];
    float z = R[6]*p.x + R[7]*p.y + R[8]*p.z + t[2];
    T[i] = make_float4(x, y, z, 0.f);
}

__global__ void icp_solve_final(const float* __restrict__ sums, float* __restrict__ out, int N) {
    if (threadIdx.x != 0 || blockIdx.x != 0) return;
    float R[3][3], t[3];
    icp_kabsch(sums, (float)N, R, t);
    for (int i = 0; i < 3; ++i) {
        out[i*4 + 0] = R[i][0];
        out[i*4 + 1] = R[i][1];
        out[i*4 + 2] = R[i][2];
        out[i*4 + 3] = t[i];
    }
}

// ---------------------------------------------------------------------------
extern "C" void kernel_launch(void* const* d_in, const int* in_sizes, int n_in,
                              void* d_out, int out_size, void* d_ws, size_t ws_size,
                              hipStream_t stream) {
    const float* p1 = (const float*)d_in[0];
    const float* p2 = (const float*)d_in[1];
    int N = in_sizes[0] / 3;                          // 8192

    float*  ws   = (float*)d_ws;
    float4* T    = (float4*)(ws);                     // temppc            [0, 4N)
    float4* P    = (float4*)(ws + 4*(size_t)N);       // p2 + bias w       [4N, 8N)
    float4* P1   = (float4*)(ws + 8*(size_t)N);       // original p1       [8N, 12N)
    int*    idx  = (int*)  (ws + 12*(size_t)N);       // [12N, 13N)
    float*  dmin =          ws + 13*(size_t)N;        // [13N, 14N)
    float*  sums =          ws + 14*(size_t)N;        // 16 floats
    float*  state=          sums + 16;                // 14 floats

    int nb = (N + 255) / 256;
    icp_prep<<<nb, 256, 0, stream>>>(p1, p2, T, P1, P, state, N);

    int knnBlocks = (N / 16 * 32 + 255) / 256;        // 1 wave per 16-row tile
    for (int it = 0; it < STEPS; ++it) {
        icp_knn<<<knnBlocks, 256, 0, stream>>>(T, P, idx, dmin, N);
        icp_reduce<<<1, 256, 0, stream>>>(T, P, idx, dmin, sums, N);
        icp_solve_step<<<1, 64, 0, stream>>>(sums, state, N);
        icp_apply<<<nb, 256, 0, stream>>>(T, state, N);
    }
    // final rigid fit: p1 -> temppc (identity pairing)
    icp_reduce<<<1, 256, 0, stream>>>(P1, T, nullptr, nullptr, sums, N);
    icp_solve_final<<<1, 64, 0, stream>>>(sums, (float*)d_out, N);
}